// FEDformer_20633022890486
// MI455X (gfx1250) — compile-verified
//
#include <hip/hip_runtime.h>

// FEDformer forward for MI455X (gfx1250, wave32).
// Exploits linearity of the moving average: trend = MA(x)@W + b,
// seasonal = (x - MA(x))@W  ->  one dual-output WMMA-f32 GEMM, no h buffer.
// Then: per-column LDS FFT (magnitudes only) -> per-batch top-64 ->
// sparse DFT coeffs -> WMMA-f32 irFFT reconstruction (+trend) ->
// LayerNorm + mean pool -> decode GEMM.

typedef float v2f __attribute__((ext_vector_type(2)));
typedef float v8f __attribute__((ext_vector_type(8)));

#define B_    32
#define L_    4096
#define IN_   64
#define D_    512
#define OUT_  96
#define LF_   2049          // rfft length L/2+1
#define MODES_ 64
#define TWO_PI 6.2831853071795864769f
#define PHASE_STEP 1.5339807878856412e-3f   // 2*pi/4096

// ---------------------------------------------------------------------------
// Kernel 1: xma = 25-tap centered moving average of x (edge/clamp padding).
// 4 rows per block; thread t owns (row t>>6, feature t&63). Fully coalesced.
// ---------------------------------------------------------------------------
__global__ __launch_bounds__(256) void xma_kernel(
    const float* __restrict__ x, float* __restrict__ xma) {
  const long row = (long)blockIdx.x * 4 + (threadIdx.x >> 6);
  const int  i   = threadIdx.x & 63;
  const int  l   = (int)(row & (L_ - 1));
  const long brow = row - l;            // b*L
  float s = 0.f;
  #pragma unroll
  for (int k = -12; k <= 12; ++k) {
    int lc = l + k; lc = lc < 0 ? 0 : (lc > L_ - 1 ? L_ - 1 : lc);
    s += x[(brow + lc) * IN_ + i];
  }
  xma[row * IN_ + i] = s * (1.f / 25.f);
}

// ---------------------------------------------------------------------------
// Kernel 2: dual-output projection via V_WMMA_F32_16X16X4_F32:
//   seas[l,d]  = (x - xma)[l,:] @ W_in[:,d]            (bias cancels)
//   trend[l,d] =  xma[l,:]      @ W_in[:,d] + b_in[d]
// M = B*L = 131072, K = 64, N = 512. One wave per 16-row tile, sweeping N.
// A layout (16x4 f32): lanes 0-15 hold K=0,1 in a.x/a.y; lanes 16-31 K=2,3.
// Both accumulators share the same B fragments of W_in.
// ---------------------------------------------------------------------------
__global__ __launch_bounds__(256) void proj2_kernel(
    const float* __restrict__ x, const float* __restrict__ xma,
    const float* __restrict__ W, const float* __restrict__ bias,
    float* __restrict__ seas, float* __restrict__ trend) {
  const int lane = threadIdx.x & 31;
  const int wv   = threadIdx.x >> 5;
  const long r0  = ((long)blockIdx.x * 8 + wv) * 16;   // M-tile base row
  const int half = lane >> 4;
  const int ln   = lane & 15;

  v2f ad[16], am[16];                   // (x - xma) and xma A-fragments
  const float* xrow = x   + (r0 + ln) * IN_;
  const float* mrow = xma + (r0 + ln) * IN_;
  #pragma unroll
  for (int j = 0; j < 16; ++j) {
    const v2f a = *(const v2f*)(xrow + 4 * j + 2 * half);
    const v2f m = *(const v2f*)(mrow + 4 * j + 2 * half);
    am[j] = m;
    ad[j].x = a.x - m.x; ad[j].y = a.y - m.y;
  }

  for (int nt = 0; nt < D_ / 16; ++nt) {
    const int n = nt * 16 + ln;
    v8f cs = {}, ct = {};
    #pragma unroll
    for (int j = 0; j < 16; ++j) {
      const int k = 4 * j + 2 * half;
      v2f bb;
      bb.x = W[(size_t)k * D_ + n];
      bb.y = W[(size_t)(k + 1) * D_ + n];
      cs = __builtin_amdgcn_wmma_f32_16x16x4_f32(false, ad[j], false, bb,
                                                 (short)0, cs, false, false);
      ct = __builtin_amdgcn_wmma_f32_16x16x4_f32(false, am[j], false, bb,
                                                 (short)0, ct, false, false);
    }
    const float bi = bias[n];
    #pragma unroll
    for (int v = 0; v < 8; ++v) {               // D row = v + 8*half, col = n
      const long row = r0 + v + 8 * half;
      seas [row * D_ + n] = cs[v];
      trend[row * D_ + n] = ct[v] + bi;
    }
  }
}

// ---------------------------------------------------------------------------
// Kernel 3: per-(b,d) 4096-pt radix-2 FFT of seasonal in LDS;
// accumulate sum-over-d of |X[b,f]| (top-k invariant to the 1/D scale).
// ---------------------------------------------------------------------------
__global__ __launch_bounds__(256) void fft_mag_kernel(
    const float* __restrict__ seas, float* __restrict__ magsum) {
  __shared__ float re[L_];
  __shared__ float im[L_];
  const int col = blockIdx.x;           // b*D + d
  const int b   = col >> 9;
  const int d   = col & (D_ - 1);
  const int t   = threadIdx.x;
  const size_t base = ((size_t)b * L_) * D_ + d;

  #pragma unroll
  for (int i = 0; i < L_ / 256; ++i) {  // bit-reversed load (12 bits)
    const int l = t + i * 256;
    const float sv = seas[base + (size_t)l * D_];
    const int r = (int)(__brev((unsigned)l) >> 20);
    re[r] = sv; im[r] = 0.f;
  }
  __syncthreads();

  for (int stage = 1; stage <= 12; ++stage) {
    const int len = 1 << stage, half = len >> 1;
    for (int p = t; p < L_ / 2; p += 256) {
      const int grp = p >> (stage - 1);
      const int pos = p & (half - 1);
      const int i = (grp << stage) + pos;
      const int j = i + half;
      const float ang = -TWO_PI * (float)pos / (float)len;
      float wi, wr; __sincosf(ang, &wi, &wr);        // e^{-i*2pi*pos/len}
      const float xr = re[j], xi = im[j];
      const float vr = xr * wr - xi * wi;
      const float vi = xr * wi + xi * wr;
      const float ur = re[i], ui = im[i];
      re[i] = ur + vr; im[i] = ui + vi;
      re[j] = ur - vr; im[j] = ui - vi;
    }
    __syncthreads();
  }

  for (int f = t; f < LF_; f += 256) {
    const float m = sqrtf(re[f] * re[f] + im[f] * im[f]);
    atomicAdd(&magsum[b * LF_ + f], m);
  }
}

// ---------------------------------------------------------------------------
// Kernel 4: per-batch top-64 frequencies by iterative LDS argmax.
// ---------------------------------------------------------------------------
__global__ __launch_bounds__(256) void topk_kernel(
    const float* __restrict__ magsum, int* __restrict__ topk) {
  __shared__ float vals[LF_];
  __shared__ float rv[256];
  __shared__ int   ri[256];
  const int b = blockIdx.x, t = threadIdx.x;
  for (int f = t; f < LF_; f += 256) vals[f] = magsum[b * LF_ + f];
  __syncthreads();
  for (int k = 0; k < MODES_; ++k) {
    float best = -1e30f; int bi = 0;
    for (int f = t; f < LF_; f += 256)
      if (vals[f] > best) { best = vals[f]; bi = f; }
    rv[t] = best; ri[t] = bi;
    __syncthreads();
    for (int s = 128; s > 0; s >>= 1) {
      if (t < s && rv[t + s] > rv[t]) { rv[t] = rv[t + s]; ri[t] = ri[t + s]; }
      __syncthreads();
    }
    if (t == 0) { topk[b * MODES_ + k] = ri[0]; vals[ri[0]] = -1e30f; }
    __syncthreads();
  }
}

// ---------------------------------------------------------------------------
// Kernel 5: DFT coefficients at kept freqs only.
// coeff[b][2m][d] = sum_l s*cos(th), coeff[b][2m+1][d] = -sum_l s*sin(th),
// th = 2*pi*f*l/L reduced exactly via (f*l) mod 4096.
// Block = one (b,m); thread t owns d = t, t+256; coalesced over l, with
// gfx1250 global_prefetch_b8 16 rows ahead.
// ---------------------------------------------------------------------------
__global__ __launch_bounds__(256) void coeff_kernel(
    const float* __restrict__ seas, const int* __restrict__ topk,
    float* __restrict__ coeff) {
  const int bm = blockIdx.x;
  const int b = bm >> 6, m = bm & (MODES_ - 1);
  const int t = threadIdx.x;
  const int f = topk[b * MODES_ + m];
  const size_t base = (size_t)b * L_ * D_;
  float ar0 = 0.f, ai0 = 0.f, ar1 = 0.f, ai1 = 0.f;
  int fl = 0;                                 // (f*l) mod 4096
  for (int l = 0; l < L_; ++l) {
    const size_t idx = base + (size_t)l * D_;
    __builtin_prefetch(&seas[idx + 16 * D_ + t], 0, 0);       // 16 rows ahead
    __builtin_prefetch(&seas[idx + 16 * D_ + t + 256], 0, 0);
    const float v0 = seas[idx + t];
    const float v1 = seas[idx + t + 256];
    float sn, cs; __sincosf(PHASE_STEP * (float)fl, &sn, &cs);
    ar0 += v0 * cs; ai0 -= v0 * sn;
    ar1 += v1 * cs; ai1 -= v1 * sn;
    fl = (fl + f) & (L_ - 1);
  }
  float* crow = coeff + ((size_t)b * 2 * MODES_ + 2 * m) * D_;
  crow[t] = ar0;       crow[t + 256] = ar1;
  crow[D_ + t] = ai0;  crow[D_ + t + 256] = ai1;
}

// ---------------------------------------------------------------------------
// Kernel 6: irFFT reconstruction + trend, as WMMA-f32 GEMM per batch:
// h2[l,d] = sum_{k=0..127} A[l,k]*coeff[k,d] + trend[l,d],
// A[l,2m] = w_m*cos(2*pi*f_m*l/L), A[l,2m+1] = -w_m*sin(...),
// w_m = 1/L for f in {0, L/2}, else 2/L.  A generated on the fly per lane.
// ---------------------------------------------------------------------------
__global__ __launch_bounds__(256) void recon_kernel(
    const float* __restrict__ coeff, const int* __restrict__ topk,
    const float* __restrict__ trend, float* __restrict__ h2) {
  const int lane = threadIdx.x & 31;
  const int wv   = threadIdx.x >> 5;
  const long tile = (long)blockIdx.x * 8 + wv;   // B*(L/16)*(D/16) tiles
  const int nt = (int)(tile & 31);
  const int lt = (int)((tile >> 5) & 255);
  const int b  = (int)(tile >> 13);
  const int half = lane >> 4;
  const int ln   = lane & 15;
  const int l = lt * 16 + ln;                    // A-row for this lane
  const int n = nt * 16 + ln;                    // B-col for this lane
  const int* fq = topk + b * MODES_;
  const float* cf = coeff + (size_t)b * 2 * MODES_ * D_;

  v8f c = {};
  #pragma unroll 4
  for (int j = 0; j < 32; ++j) {                 // K = 128 in steps of 4
    const int m0 = 2 * j + half;                 // mode for this lane's pair
    const int f  = fq[m0];
    const float wm = (f == 0 || f == L_ / 2) ? (1.f / L_) : (2.f / L_);
    const int p = (f * l) & (L_ - 1);
    float sn, cs; __sincosf(PHASE_STEP * (float)p, &sn, &cs);
    v2f a; a.x = wm * cs; a.y = -wm * sn;        // K even / K odd elements
    const int k0 = 4 * j + 2 * half;
    v2f bb;
    bb.x = cf[(size_t)k0 * D_ + n];
    bb.y = cf[(size_t)(k0 + 1) * D_ + n];
    c = __builtin_amdgcn_wmma_f32_16x16x4_f32(false, a, false, bb,
                                              (short)0, c, false, false);
  }
  const size_t base = (size_t)b * L_ * D_;
  const int l0 = lt * 16;
  #pragma unroll
  for (int v = 0; v < 8; ++v) {
    const size_t idx = base + (size_t)(l0 + v + 8 * half) * D_ + n;
    h2[idx] = c[v] + trend[idx];
  }
}

// ---------------------------------------------------------------------------
// Kernel 7: LayerNorm over D + mean-pool over L (accumulated via atomics).
// Block handles 16 consecutive rows of one batch; thread t owns d=t, t+256.
// ---------------------------------------------------------------------------
__global__ __launch_bounds__(256) void ln_pool_kernel(
    const float* __restrict__ h2, const float* __restrict__ gamma,
    const float* __restrict__ beta, float* __restrict__ rep) {
  __shared__ float red[256];
  const int t = threadIdx.x;
  const long row0 = (long)blockIdx.x * 16;
  const int b = (int)(row0 >> 12);
  const float g0 = gamma[t], g1 = gamma[t + 256];
  const float e0 = beta[t],  e1 = beta[t + 256];
  float acc0 = 0.f, acc1 = 0.f;
  for (int i = 0; i < 16; ++i) {
    const long row = row0 + i;
    const float v0 = h2[row * D_ + t];
    const float v1 = h2[row * D_ + t + 256];
    red[t] = v0 + v1; __syncthreads();
    for (int s = 128; s > 0; s >>= 1) { if (t < s) red[t] += red[t + s]; __syncthreads(); }
    const float mu = red[0] * (1.f / D_); __syncthreads();
    const float d0 = v0 - mu, d1 = v1 - mu;
    red[t] = d0 * d0 + d1 * d1; __syncthreads();
    for (int s = 128; s > 0; s >>= 1) { if (t < s) red[t] += red[t + s]; __syncthreads(); }
    const float inv = rsqrtf(red[0] * (1.f / D_) + 1e-5f); __syncthreads();
    acc0 += d0 * inv * g0 + e0;
    acc1 += d1 * inv * g1 + e1;
  }
  atomicAdd(&rep[b * D_ + t], acc0);
  atomicAdd(&rep[b * D_ + t + 256], acc1);
}

// ---------------------------------------------------------------------------
// Kernel 8: out[b,o] = (rep[b,:] / L) @ W_dec[:,o] + b_dec[o]
// ---------------------------------------------------------------------------
__global__ void dec_kernel(const float* __restrict__ rep,
                           const float* __restrict__ Wd,
                           const float* __restrict__ bd,
                           float* __restrict__ out) {
  const int b = blockIdx.x, o = threadIdx.x;    // OUT_ threads
  float acc = 0.f;
  for (int d = 0; d < D_; ++d) acc += rep[b * D_ + d] * Wd[d * OUT_ + o];
  out[b * OUT_ + o] = acc * (1.f / L_) + bd[o];
}

// ---------------------------------------------------------------------------
extern "C" void kernel_launch(void* const* d_in, const int* in_sizes, int n_in,
                              void* d_out, int out_size, void* d_ws, size_t ws_size,
                              hipStream_t stream) {
  (void)in_sizes; (void)n_in; (void)out_size; (void)ws_size;
  const float* x     = (const float*)d_in[0];
  const float* W_in  = (const float*)d_in[1];
  const float* b_in  = (const float*)d_in[2];
  const float* gamma = (const float*)d_in[3];
  const float* beta  = (const float*)d_in[4];
  const float* W_dec = (const float*)d_in[5];
  const float* b_dec = (const float*)d_in[6];
  float* out = (float*)d_out;

  const size_t NH  = (size_t)B_ * L_ * D_;      // 67,108,864 floats (256 MB)
  const size_t NXM = (size_t)B_ * L_ * IN_;     //  8,388,608 floats (32 MB)
  float* seas  = (float*)d_ws;                  // 256 MB (reused as h2)
  float* trend = seas + NH;                     // 256 MB
  float* xma   = trend + NH;                    // 32 MB
  float* mag   = xma + NXM;                     // 32*2049 floats
  int*   topk  = (int*)(mag + (size_t)B_ * LF_);
  float* coeff = (float*)(topk + B_ * MODES_);  // 32*128*512 floats (8 MB)
  float* rep   = coeff + (size_t)B_ * 2 * MODES_ * D_;   // 32*512 floats

  hipMemsetAsync(mag, 0, (size_t)B_ * LF_ * sizeof(float), stream);
  hipMemsetAsync(rep, 0, (size_t)B_ * D_ * sizeof(float), stream);

  xma_kernel    <<<B_ * L_ / 4,       256, 0, stream>>>(x, xma);
  proj2_kernel  <<<(B_ * L_ / 16) / 8, 256, 0, stream>>>(x, xma, W_in, b_in,
                                                         seas, trend);
  fft_mag_kernel<<<B_ * D_,           256, 0, stream>>>(seas, mag);
  topk_kernel   <<<B_,                256, 0, stream>>>(mag, topk);
  coeff_kernel  <<<B_ * MODES_,       256, 0, stream>>>(seas, topk, coeff);
  recon_kernel  <<<(B_ * (L_ / 16) * (D_ / 16)) / 8, 256, 0, stream>>>(
      coeff, topk, trend, /*h2 overwrites seas*/ seas);
  ln_pool_kernel<<<(B_ * L_) / 16,    256, 0, stream>>>(seas, gamma, beta, rep);
  dec_kernel    <<<B_, OUT_,            0, stream>>>(rep, W_dec, b_dec, out);
}